// PeakExtractor_69063074120418
// MI455X (gfx1250) — compile-verified
//
#include <hip/hip_runtime.h>
#include <hip/hip_bf16.h>
#include <stdint.h>

// ---------------------------------------------------------------------------
// Peak extractor: 5x5 NMS max-pool + per-batch top-3 over [bs, 5*256, 256].
// Memory-bound (83.9 MB in, ~4 KB out -> ~3.6 us floor @ 23.3 TB/s).
// CDNA5 path: async global->LDS tile staging (ASYNCcnt) for the stencil.
// ---------------------------------------------------------------------------

#define W        256          // heatmap width
#define NVIEW    5
#define HV       256          // per-view height
#define NR       (NVIEW * HV) // 1280 flattened rows per batch
#define TR       16           // output rows per block
#define THALO    (TR + 4)     // tile rows incl. 2-row halo each side
#define NEG_FILL -1.0e9f

typedef unsigned long long ull;

// ---- CDNA5 async global->LDS copy (GLOBAL_LOAD_ASYNC_TO_LDS_B128) ---------
__device__ __forceinline__ void async_load_b128_to_lds(unsigned lds_byte_off,
                                                       const float* gptr) {
    // VDST = LDS byte address (VGPR), VADDR = 64-bit global address (VGPR pair)
    asm volatile("global_load_async_to_lds_b128 %0, %1, off"
                 :: "v"(lds_byte_off), "v"(gptr)
                 : "memory");
}

__device__ __forceinline__ void wait_asynccnt_zero() {
#if __has_builtin(__builtin_amdgcn_s_wait_asynccnt)
    __builtin_amdgcn_s_wait_asynccnt(0);
#else
    asm volatile("s_wait_asynccnt 0" ::: "memory");
#endif
}

// Order-preserving f32 -> u32 map (monotonic: a<b  <=>  map(a)<map(b))
__device__ __forceinline__ unsigned mono_f32(float f) {
    unsigned b = __float_as_uint(f);
    return (b & 0x80000000u) ? ~b : (b | 0x80000000u);
}

// Merge two descending sorted triples -> top-3 descending
__device__ __forceinline__ void merge3(ull* dst, const ull* a, const ull* b) {
    ull av0=a[0], av1=a[1], av2=a[2];
    ull bv0=b[0], bv1=b[1], bv2=b[2];
    ull av[3] = {av0, av1, av2};
    ull bv[3] = {bv0, bv1, bv2};
    ull out[3];
    int ia = 0, ib = 0;
    #pragma unroll
    for (int k = 0; k < 3; ++k) {
        if (av[ia] >= bv[ib]) out[k] = av[ia++];
        else                  out[k] = bv[ib++];
    }
    dst[0] = out[0]; dst[1] = out[1]; dst[2] = out[2];
}

// ---------------------------------------------------------------------------
// Stage 1: one block per (tile-of-16-rows, batch). Separable 5x5 max via LDS,
// per-block top-3 candidate keys written to workspace.
// ---------------------------------------------------------------------------
__global__ __launch_bounds__(256)
void peaks_stage1(const float* __restrict__ hm,
                  ull* __restrict__ block_top,
                  int tiles_per_batch) {
    __shared__ float tile[THALO][W];   // 20 KB
    __shared__ float hmax[THALO][W];   // 20 KB
    __shared__ ull   keys[256 * 3];    // 6 KB

    const int tile_id = blockIdx.x;
    const int b       = blockIdx.y;
    const int t       = threadIdx.x;
    const int row0    = tile_id * TR;              // first output row of tile
    const size_t base = (size_t)b * (size_t)(NR * W);
    const float NEGINF = -__builtin_inff();

    // ---- stage tile (with halo) into LDS: async B128 per 4-float slot -----
    const int NSLOT = THALO * (W / 4);             // 1280 float4 slots
    for (int s = t; s < NSLOT; s += 256) {
        const int r    = s >> 6;                   // slot / 64
        const int c4   = (s & 63) << 2;            // 4-float column start
        const int grow = row0 - 2 + r;             // global row in [−2, 1282)
        if (grow >= 0 && grow < NR) {
            const unsigned lds_off =
                (unsigned)(uintptr_t)(void*)&tile[r][c4]; // low32 = LDS offset
            const float* g = hm + base + (size_t)grow * W + c4;
            async_load_b128_to_lds(lds_off, g);
        } else {
            // -inf padding rows outside the 1280-row flattened image
            float4 v; v.x = NEGINF; v.y = NEGINF; v.z = NEGINF; v.w = NEGINF;
            *(float4*)&tile[r][c4] = v;
        }
    }
    wait_asynccnt_zero();     // each wave drains its own async copies
    __syncthreads();          // publish LDS tile across all 8 waves

    // ---- horizontal 5-max (column edges behave as -inf pad) ---------------
    {
        const int c = t;
        #pragma unroll 4
        for (int r = 0; r < THALO; ++r) {
            float m = tile[r][c];
            if (c >= 1)     m = fmaxf(m, tile[r][c - 1]);
            if (c >= 2)     m = fmaxf(m, tile[r][c - 2]);
            if (c <= W - 2) m = fmaxf(m, tile[r][c + 1]);
            if (c <= W - 3) m = fmaxf(m, tile[r][c + 2]);
            hmax[r][c] = m;
        }
    }
    __syncthreads();

    // ---- vertical 5-max, peak test, per-thread top-3 ----------------------
    // Every pixel is a candidate (non-peaks scored NEG_FILL) so the global
    // top-3 exactly matches jax.lax.top_k incl. tie-breaking on low index.
    ull k0 = 0, k1 = 0, k2 = 0;   // 0 < any real key (monotone map > 0)
    {
        const int c = t;
        #pragma unroll 4
        for (int j = 0; j < TR; ++j) {
            const float center = tile[j + 2][c];
            float wmax = hmax[j][c];
            wmax = fmaxf(wmax, hmax[j + 1][c]);
            wmax = fmaxf(wmax, hmax[j + 2][c]);
            wmax = fmaxf(wmax, hmax[j + 3][c]);
            wmax = fmaxf(wmax, hmax[j + 4][c]);
            const bool  peak  = (center == wmax);
            const float score = peak ? center : NEG_FILL;
            const unsigned idx = (unsigned)((row0 + j) * W + c);
            const ull key = ((ull)mono_f32(score) << 32) | (unsigned)(~idx);
            if      (key > k0) { k2 = k1; k1 = k0; k0 = key; }
            else if (key > k1) { k2 = k1; k1 = key; }
            else if (key > k2) { k2 = key; }
        }
    }
    keys[t * 3 + 0] = k0;
    keys[t * 3 + 1] = k1;
    keys[t * 3 + 2] = k2;
    __syncthreads();

    // ---- block-level top-3: log2(256) sorted-triple merges ----------------
    for (int s = 128; s >= 1; s >>= 1) {
        if (t < s) {
            ull a[3] = {keys[t * 3], keys[t * 3 + 1], keys[t * 3 + 2]};
            ull bq[3] = {keys[(t + s) * 3], keys[(t + s) * 3 + 1],
                         keys[(t + s) * 3 + 2]};
            ull r[3];
            merge3(r, a, bq);
            keys[t * 3] = r[0]; keys[t * 3 + 1] = r[1]; keys[t * 3 + 2] = r[2];
        }
        __syncthreads();
    }

    if (t == 0) {
        ull* dst = block_top + ((size_t)b * tiles_per_batch + tile_id) * 3;
        dst[0] = keys[0]; dst[1] = keys[1]; dst[2] = keys[2];
    }
}

// ---------------------------------------------------------------------------
// Stage 2: one thread per batch. Reduce 80*3 candidate keys -> top-3, decode.
// Output layout (floats): positions [bs,3,3] | scores [bs,3] | mask [bs,3].
// ---------------------------------------------------------------------------
__global__ void peaks_stage2(const ull* __restrict__ block_top,
                             float* __restrict__ out,
                             int tiles_per_batch, int bs) {
    const int b = threadIdx.x + blockIdx.x * blockDim.x;
    if (b >= bs) return;

    ull k0 = 0, k1 = 0, k2 = 0;
    const ull* p = block_top + (size_t)b * tiles_per_batch * 3;
    const int n = tiles_per_batch * 3;
    for (int i = 0; i < n; ++i) {
        const ull key = p[i];
        if      (key > k0) { k2 = k1; k1 = k0; k0 = key; }
        else if (key > k1) { k2 = k1; k1 = key; }
        else if (key > k2) { k2 = key; }
    }

    const int pos_off   = 0;
    const int score_off = bs * 9;
    const int mask_off  = bs * 9 + bs * 3;

    ull ks[3] = {k0, k1, k2};
    #pragma unroll
    for (int k = 0; k < 3; ++k) {
        const unsigned mono = (unsigned)(ks[k] >> 32);
        const unsigned bits = (mono & 0x80000000u) ? (mono & 0x7fffffffu)
                                                   : ~mono;
        const float score = __uint_as_float(bits);
        const unsigned idx = ~(unsigned)(ks[k] & 0xffffffffu);
        const unsigned view = idx >> 16;          // / (256*256)
        const unsigned rem  = idx & 0xffffu;
        const unsigned row  = rem >> 8;
        const unsigned col  = rem & 255u;
        out[pos_off + b * 9 + k * 3 + 0] = (float)view;
        out[pos_off + b * 9 + k * 3 + 1] = (float)row;
        out[pos_off + b * 9 + k * 3 + 2] = (float)col;
        out[score_off + b * 3 + k] = score;
        out[mask_off  + b * 3 + k] = (score > -1.0e30f) ? 1.0f : 0.0f;
    }
}

// ---------------------------------------------------------------------------
extern "C" void kernel_launch(void* const* d_in, const int* in_sizes, int n_in,
                              void* d_out, int out_size, void* d_ws,
                              size_t ws_size, hipStream_t stream) {
    const float* hm = (const float*)d_in[0];
    float* out = (float*)d_out;

    const int per_batch = NVIEW * HV * W;              // 327,680
    const int bs = in_sizes[0] / per_batch;            // 64
    const int tiles_per_batch = NR / TR;               // 80

    ull* block_top = (ull*)d_ws;                       // bs*80*3 u64 = 120 KB

    peaks_stage1<<<dim3(tiles_per_batch, bs), 256, 0, stream>>>(
        hm, block_top, tiles_per_batch);
    peaks_stage2<<<1, bs, 0, stream>>>(block_top, out, tiles_per_batch, bs);
}